// MultiHeadGCNGATVGAE_3513283248866
// MI455X (gfx1250) — compile-verified
//
#include <hip/hip_runtime.h>
#include <math.h>

typedef float v2f __attribute__((ext_vector_type(2)));
typedef float v8f __attribute__((ext_vector_type(8)));

#define DEV_INLINE __device__ __forceinline__

constexpr int NN    = 8192;     // nodes
constexpr int FIN   = 512;
constexpr int H1    = 128;
constexpr int H2    = 64;
constexpr int HEADS = 3;
constexpr int NE    = 262144;   // edges (self loops handled analytically)

DEV_INLINE float leaky(float x) { return x > 0.f ? x : 0.2f * x; }

// monotonic float<->uint key for atomicMax on floats (handles negatives, -0<+0)
DEV_INLINE unsigned f2key(float f) {
  unsigned b = __float_as_uint(f);
  return (b & 0x80000000u) ? ~b : (b | 0x80000000u);
}
DEV_INLINE float key2f(unsigned k) {
  unsigned b = (k & 0x80000000u) ? (k ^ 0x80000000u) : ~k;
  return __uint_as_float(b);
}

// ---------------------------------------------------------------- degree
__global__ void k_deg_init(float* deg, int n) {
  int i = blockIdx.x * blockDim.x + threadIdx.x;
  if (i < n) deg[i] = 1.0f;  // self loop
}
__global__ void k_deg_edges(const int* __restrict__ dst, float* deg, int e) {
  int i = blockIdx.x * blockDim.x + threadIdx.x;
  if (i < e) atomicAdd(&deg[dst[i]], 1.0f);
}
__global__ void k_deg_fin(float* deg, int n) {
  int i = blockIdx.x * blockDim.x + threadIdx.x;
  if (i < n) deg[i] = rsqrtf(deg[i]);  // deg >= 1 always (self loops)
}

// ------------------------------------------------- dense GEMM via WMMA f32
// C[M,Nc] = A[M,K] @ B[K,Nc], row-major fp32. One wave per 16x64 strip
// (4 accumulators; A-fragment reused across 4 column tiles). Nc % 64 == 0.
__global__ __launch_bounds__(256) void k_gemm_wmma(
    const float* __restrict__ A, const float* __restrict__ B,
    float* __restrict__ C, int M, int Nc, int K) {
  const int lane = threadIdx.x & 31;
  const int wave = blockIdx.x * (blockDim.x >> 5) + (threadIdx.x >> 5);
  const int grpN = Nc >> 6;                 // groups of 64 columns
  const int strips = (M >> 4) * grpN;
  if (wave >= strips) return;               // whole-wave exit: EXEC stays all-1s
  const int ti = wave / grpN, tg = wave % grpN;
  const int r = lane & 15, hi = lane >> 4;  // hi in {0,1}
  const float* Arow = A + (size_t)(ti * 16 + r) * K;
  const float* Bcol = B + (size_t)(tg * 64 + r);
  v8f acc[4] = {v8f{}, v8f{}, v8f{}, v8f{}};
  for (int k0 = 0; k0 < K; k0 += 4) {
    const int ka = k0 + hi * 2;
    v2f a;
    a[0] = Arow[ka];
    a[1] = Arow[ka + 1];
    const float* bp0 = Bcol + (size_t)ka * Nc;
    const float* bp1 = Bcol + (size_t)(ka + 1) * Nc;
#pragma unroll
    for (int q = 0; q < 4; ++q) {
      v2f b;
      b[0] = bp0[q * 16];
      b[1] = bp1[q * 16];
      acc[q] = __builtin_amdgcn_wmma_f32_16x16x4_f32(false, a, false, b,
                                                     (short)0, acc[q], false, false);
    }
  }
  const int m0 = ti * 16 + hi * 8;
#pragma unroll
  for (int q = 0; q < 4; ++q) {
    float* Cp = C + (size_t)(tg * 64 + q * 16 + r);
#pragma unroll
    for (int v = 0; v < 8; ++v) Cp[(size_t)(m0 + v) * Nc] = acc[q][v];
  }
}

// --------------------------------------- decoder: sigmoid(Z @ Z^T), Z [n,k]
// One wave per 16x64 output strip: 4 accumulators, A-fragment reused 4x,
// cutting L2 read traffic ~40% vs one-tile-per-wave (store-BW bound kernel).
__global__ __launch_bounds__(256) void k_decoder(
    const float* __restrict__ Z, float* __restrict__ out, int n, int k) {
  const int lane = threadIdx.x & 31;
  const int wave = blockIdx.x * (blockDim.x >> 5) + (threadIdx.x >> 5);
  const int tilesI = n >> 4;
  const int grpJ   = n >> 6;
  if (wave >= tilesI * grpJ) return;
  const int ti = wave / grpJ, tg = wave % grpJ;
  const int r = lane & 15, hi = lane >> 4;
  const float* Arow = Z + (size_t)(ti * 16 + r) * k;
  const float* Brow = Z + (size_t)(tg * 64 + r) * k;  // B = Z^T -> read Z rows
  v8f acc[4] = {v8f{}, v8f{}, v8f{}, v8f{}};
  for (int k0 = 0; k0 < k; k0 += 4) {
    const int ka = k0 + hi * 2;
    v2f a;
    a[0] = Arow[ka];
    a[1] = Arow[ka + 1];
#pragma unroll
    for (int q = 0; q < 4; ++q) {
      const float* bp = Brow + (size_t)(q * 16) * k;
      v2f b;
      b[0] = bp[ka];
      b[1] = bp[ka + 1];
      acc[q] = __builtin_amdgcn_wmma_f32_16x16x4_f32(false, a, false, b,
                                                     (short)0, acc[q], false, false);
    }
  }
  const int m0 = ti * 16 + hi * 8;
#pragma unroll
  for (int q = 0; q < 4; ++q) {
    float* Op = out + (size_t)(tg * 64 + q * 16 + r);
#pragma unroll
    for (int v = 0; v < 8; ++v) {
      float x = acc[q][v];
      Op[(size_t)(m0 + v) * n] = 1.0f / (1.0f + expf(-x));
    }
  }
}

// ---------------------------------------------------------------- GCN agg
__global__ void k_gcn_agg_init(const float* __restrict__ t0, const float* __restrict__ dinv,
                               const float* __restrict__ bias, float* h, int n, int F) {
  long long i = (long long)blockIdx.x * blockDim.x + threadIdx.x;
  if (i >= (long long)n * F) return;
  int node = (int)(i / F), f = (int)(i % F);
  float dv = dinv[node];
  h[i] = t0[i] * dv * dv + bias[f];  // self loop + bias
}
__global__ void k_gcn_agg_edges(const float* __restrict__ t0, const float* __restrict__ dinv,
                                const int* __restrict__ src, const int* __restrict__ dst,
                                float* h, int e, int F) {
  long long i = (long long)blockIdx.x * blockDim.x + threadIdx.x;
  if (i >= (long long)e * F) return;
  int ed = (int)(i / F), f = (int)(i % F);
  int s = src[ed], d = dst[ed];
  atomicAdd(&h[(size_t)d * F + f], t0[(size_t)s * F + f] * dinv[s] * dinv[d]);
}

// ------------------------------------------------------------- elementwise
__global__ void k_relu(float* x, long long cnt) {
  long long i = (long long)blockIdx.x * blockDim.x + threadIdx.x;
  if (i < cnt) x[i] = fmaxf(x[i], 0.f);
}
__global__ void k_bias_relu(float* x, const float* __restrict__ b, int n, int F) {
  long long i = (long long)blockIdx.x * blockDim.x + threadIdx.x;
  if (i >= (long long)n * F) return;
  x[i] = fmaxf(x[i] + b[i % F], 0.f);
}

// ---------------------------------------------------------------- GAT
__global__ void k_att(const float* __restrict__ h, const float* __restrict__ avec,
                      float* out, int n, int heads, int D) {
  int i = blockIdx.x * blockDim.x + threadIdx.x;  // n*heads
  if (i >= n * heads) return;
  int node = i / heads, hh = i % heads;
  const float* hp = h + (size_t)node * heads * D + hh * D;
  const float* ap = avec + hh * D;
  float s = 0.f;
  for (int d = 0; d < D; ++d) s += hp[d] * ap[d];
  out[i] = s;
}
__global__ void k_m_init(const float* __restrict__ as_, const float* __restrict__ ad_,
                         unsigned* mkey, int cnt) {
  int i = blockIdx.x * blockDim.x + threadIdx.x;
  if (i < cnt) mkey[i] = f2key(leaky(as_[i] + ad_[i]));  // self loop logit
}
__global__ void k_m_edges(const float* __restrict__ as_, const float* __restrict__ ad_,
                          const int* __restrict__ src, const int* __restrict__ dst,
                          unsigned* mkey, int e, int heads) {
  int i = blockIdx.x * blockDim.x + threadIdx.x;
  if (i >= e * heads) return;
  int ed = i / heads, hh = i % heads;
  int s = src[ed], d = dst[ed];
  atomicMax(&mkey[d * heads + hh], f2key(leaky(as_[s * heads + hh] + ad_[d * heads + hh])));
}
__global__ void k_den_init(const float* __restrict__ as_, const float* __restrict__ ad_,
                           const unsigned* __restrict__ mkey, float* den, int cnt) {
  int i = blockIdx.x * blockDim.x + threadIdx.x;
  if (i < cnt) den[i] = expf(leaky(as_[i] + ad_[i]) - key2f(mkey[i]));
}
__global__ void k_den_edges(const float* __restrict__ as_, const float* __restrict__ ad_,
                            const int* __restrict__ src, const int* __restrict__ dst,
                            const unsigned* __restrict__ mkey, float* den, int e, int heads) {
  int i = blockIdx.x * blockDim.x + threadIdx.x;
  if (i >= e * heads) return;
  int ed = i / heads, hh = i % heads;
  int s = src[ed], d = dst[ed];
  int di = d * heads + hh;
  atomicAdd(&den[di], expf(leaky(as_[s * heads + hh] + ad_[di]) - key2f(mkey[di])));
}
__global__ void k_gat_agg_init(const float* __restrict__ hpre, const float* __restrict__ as_,
                               const float* __restrict__ ad_, const unsigned* __restrict__ mkey,
                               const float* __restrict__ den, float* out, int n, int heads, int D) {
  int i = blockIdx.x * blockDim.x + threadIdx.x;  // n*heads
  if (i >= n * heads) return;
  int node = i / heads, hh = i % heads;
  float w = expf(leaky(as_[i] + ad_[i]) - key2f(mkey[i])) / den[i];
  const float* hp = hpre + (size_t)node * heads * D + hh * D;
  float* op = out + (size_t)node * heads * D + hh * D;
  for (int d = 0; d < D; ++d) op[d] = w * hp[d];
}
__global__ void k_gat_agg_edges(const float* __restrict__ hpre, const float* __restrict__ as_,
                                const float* __restrict__ ad_, const unsigned* __restrict__ mkey,
                                const float* __restrict__ den, const int* __restrict__ src,
                                const int* __restrict__ dst, float* out, int e, int heads, int D) {
  int i = blockIdx.x * blockDim.x + threadIdx.x;  // e*heads
  if (i >= e * heads) return;
  int ed = i / heads, hh = i % heads;
  int s = src[ed], d = dst[ed];
  int di = d * heads + hh;
  float w = expf(leaky(as_[s * heads + hh] + ad_[di]) - key2f(mkey[di])) / den[di];
  const float* hp = hpre + (size_t)s * heads * D + hh * D;
  float* op = out + (size_t)d * heads * D + hh * D;
  for (int dd = 0; dd < D; ++dd) atomicAdd(&op[dd], w * hp[dd]);
}
__global__ void k_head_mean_bias(const float* __restrict__ out2, const float* __restrict__ b,
                                 float* mu, int n, int heads, int D) {
  int i = blockIdx.x * blockDim.x + threadIdx.x;  // n*D
  if (i >= n * D) return;
  int node = i / D, d = i % D;
  float s = 0.f;
  for (int h = 0; h < heads; ++h) s += out2[(size_t)node * heads * D + h * D + d];
  mu[i] = s / (float)heads + b[d];
}

// -------------------------------------------------------------- combine
__global__ void k_combine(const float* __restrict__ mu, const float* __restrict__ lv,
                          const float* __restrict__ eps, float* mu_acc, float* lv_acc,
                          float* z_acc, long long cnt, int first) {
  long long i = (long long)blockIdx.x * blockDim.x + threadIdx.x;
  if (i >= cnt) return;
  float m = mu[i], l = lv[i];
  float z = m + eps[i] * expf(l);
  if (first) { mu_acc[i] = m; lv_acc[i] = l; z_acc[i] = z; }
  else {
    mu_acc[i] = fmaxf(mu_acc[i], m);
    lv_acc[i] = fmaxf(lv_acc[i], l);
    z_acc[i]  = fmaxf(z_acc[i], z);
  }
}
__global__ void k_tail(const float* __restrict__ mu_acc, const float* __restrict__ lv_acc,
                       float* out, int cnt) {
  int i = blockIdx.x * blockDim.x + threadIdx.x;
  if (i >= cnt) return;
  size_t base = (size_t)NN * NN;
  out[base + i] = mu_acc[i];
  out[base + cnt + i] = lv_acc[i];
}

// ----------------------------------------------------------------- host
static inline void gemm(const float* A, const float* B, float* C,
                        int M, int Nc, int K, hipStream_t s) {
  int strips = (M / 16) * (Nc / 64);       // one wave per 16x64 strip
  int blocks = (strips + 7) / 8;
  k_gemm_wmma<<<dim3(blocks), dim3(256), 0, s>>>(A, B, C, M, Nc, K);
}
static inline dim3 g1(long long cnt) { return dim3((unsigned)((cnt + 255) / 256)); }

extern "C" void kernel_launch(void* const* d_in, const int* in_sizes, int n_in,
                              void* d_out, int out_size, void* d_ws, size_t ws_size,
                              hipStream_t stream) {
  const float* x       = (const float*)d_in[0];
  const int*   ei      = (const int*)d_in[1];
  const int*   src     = ei;
  const int*   dst     = ei + NE;
  const float* gcn_W1  = (const float*)d_in[2];
  const float* gcn_b1  = (const float*)d_in[3];
  const float* gcn_Wmu = (const float*)d_in[4];
  const float* gcn_bmu = (const float*)d_in[5];
  const float* gcn_Wlv = (const float*)d_in[6];
  const float* gcn_blv = (const float*)d_in[7];
  const float* gat_W1  = (const float*)d_in[8];
  const float* gat_as1 = (const float*)d_in[9];
  const float* gat_ad1 = (const float*)d_in[10];
  const float* gat_b1  = (const float*)d_in[11];
  const float* gat_Wmu = (const float*)d_in[12];
  const float* gat_asmu= (const float*)d_in[13];
  const float* gat_admu= (const float*)d_in[14];
  const float* gat_bmu = (const float*)d_in[15];
  const float* gat_Wlv = (const float*)d_in[16];
  const float* gat_aslv= (const float*)d_in[17];
  const float* gat_adlv= (const float*)d_in[18];
  const float* gat_blv = (const float*)d_in[19];
  const float* eps_gcn = (const float*)d_in[20];
  const float* eps_gat = (const float*)d_in[21];
  float* out = (float*)d_out;

  // workspace layout (floats)
  float* w = (float*)d_ws;
  float* dinv   = w; w += NN;
  float* t0     = w; w += (size_t)NN * 384;   // layer-1 XW (max width 384)
  float* hbuf   = w; w += (size_t)NN * 384;   // layer-1 aggregated / hidden
  float* a_s    = w; w += NN * HEADS;
  float* a_d    = w; w += NN * HEADS;
  unsigned* mkey= (unsigned*)w; w += NN * HEADS;
  float* den    = w; w += NN * HEADS;
  float* t1     = w; w += (size_t)NN * 192;   // layer-2 XW (max width 192)
  float* out2   = w; w += (size_t)NN * 192;   // layer-2 aggregated (GAT pre-mean)
  float* mu     = w; w += NN * H2;
  float* lv     = w; w += NN * H2;
  float* mu_acc = w; w += NN * H2;
  float* lv_acc = w; w += NN * H2;
  float* z_acc  = w; w += NN * H2;

  // degrees -> dinv (shared by all branches)
  k_deg_init<<<g1(NN), 256, 0, stream>>>(dinv, NN);
  k_deg_edges<<<g1(NE), 256, 0, stream>>>(dst, dinv, NE);
  k_deg_fin<<<g1(NN), 256, 0, stream>>>(dinv, NN);

  // ----------------- GCN branches -----------------
  for (int i = 0; i < 2; ++i) {
    gemm(x, gcn_W1 + (size_t)i * FIN * H1, t0, NN, H1, FIN, stream);
    k_gcn_agg_init<<<g1((long long)NN * H1), 256, 0, stream>>>(t0, dinv, gcn_b1 + i * H1, hbuf, NN, H1);
    k_gcn_agg_edges<<<g1((long long)NE * H1), 256, 0, stream>>>(t0, dinv, src, dst, hbuf, NE, H1);
    k_relu<<<g1((long long)NN * H1), 256, 0, stream>>>(hbuf, (long long)NN * H1);
    // mu
    gemm(hbuf, gcn_Wmu + (size_t)i * H1 * H2, t1, NN, H2, H1, stream);
    k_gcn_agg_init<<<g1((long long)NN * H2), 256, 0, stream>>>(t1, dinv, gcn_bmu + i * H2, mu, NN, H2);
    k_gcn_agg_edges<<<g1((long long)NE * H2), 256, 0, stream>>>(t1, dinv, src, dst, mu, NE, H2);
    // lv
    gemm(hbuf, gcn_Wlv + (size_t)i * H1 * H2, t1, NN, H2, H1, stream);
    k_gcn_agg_init<<<g1((long long)NN * H2), 256, 0, stream>>>(t1, dinv, gcn_blv + i * H2, lv, NN, H2);
    k_gcn_agg_edges<<<g1((long long)NE * H2), 256, 0, stream>>>(t1, dinv, src, dst, lv, NE, H2);
    k_combine<<<g1((long long)NN * H2), 256, 0, stream>>>(mu, lv, eps_gcn + (size_t)i * NN * H2,
                                                          mu_acc, lv_acc, z_acc,
                                                          (long long)NN * H2, i == 0 ? 1 : 0);
  }

  // ----------------- GAT branches -----------------
  const int D1 = H1;                 // per-head dim layer 1 (128)
  const int W1c = HEADS * H1;        // 384
  const int W2c = HEADS * H2;        // 192
  for (int i = 0; i < 2; ++i) {
    // layer 1 (concat=True)
    gemm(x, gat_W1 + (size_t)i * FIN * W1c, t0, NN, W1c, FIN, stream);
    k_att<<<g1(NN * HEADS), 256, 0, stream>>>(t0, gat_as1 + i * W1c, a_s, NN, HEADS, D1);
    k_att<<<g1(NN * HEADS), 256, 0, stream>>>(t0, gat_ad1 + i * W1c, a_d, NN, HEADS, D1);
    k_m_init<<<g1(NN * HEADS), 256, 0, stream>>>(a_s, a_d, mkey, NN * HEADS);
    k_m_edges<<<g1(NE * HEADS), 256, 0, stream>>>(a_s, a_d, src, dst, mkey, NE, HEADS);
    k_den_init<<<g1(NN * HEADS), 256, 0, stream>>>(a_s, a_d, mkey, den, NN * HEADS);
    k_den_edges<<<g1(NE * HEADS), 256, 0, stream>>>(a_s, a_d, src, dst, mkey, den, NE, HEADS);
    k_gat_agg_init<<<g1(NN * HEADS), 256, 0, stream>>>(t0, a_s, a_d, mkey, den, hbuf, NN, HEADS, D1);
    k_gat_agg_edges<<<g1(NE * HEADS), 256, 0, stream>>>(t0, a_s, a_d, mkey, den, src, dst, hbuf, NE, HEADS, D1);
    k_bias_relu<<<g1((long long)NN * W1c), 256, 0, stream>>>(hbuf, gat_b1 + i * W1c, NN, W1c);

    // layer 2: mu (concat=False -> head mean + bias)
    gemm(hbuf, gat_Wmu + (size_t)i * W1c * W2c, t1, NN, W2c, W1c, stream);
    k_att<<<g1(NN * HEADS), 256, 0, stream>>>(t1, gat_asmu + i * W2c, a_s, NN, HEADS, H2);
    k_att<<<g1(NN * HEADS), 256, 0, stream>>>(t1, gat_admu + i * W2c, a_d, NN, HEADS, H2);
    k_m_init<<<g1(NN * HEADS), 256, 0, stream>>>(a_s, a_d, mkey, NN * HEADS);
    k_m_edges<<<g1(NE * HEADS), 256, 0, stream>>>(a_s, a_d, src, dst, mkey, NE, HEADS);
    k_den_init<<<g1(NN * HEADS), 256, 0, stream>>>(a_s, a_d, mkey, den, NN * HEADS);
    k_den_edges<<<g1(NE * HEADS), 256, 0, stream>>>(a_s, a_d, src, dst, mkey, den, NE, HEADS);
    k_gat_agg_init<<<g1(NN * HEADS), 256, 0, stream>>>(t1, a_s, a_d, mkey, den, out2, NN, HEADS, H2);
    k_gat_agg_edges<<<g1(NE * HEADS), 256, 0, stream>>>(t1, a_s, a_d, mkey, den, src, dst, out2, NE, HEADS, H2);
    k_head_mean_bias<<<g1(NN * H2), 256, 0, stream>>>(out2, gat_bmu + i * H2, mu, NN, HEADS, H2);

    // layer 2: lv
    gemm(hbuf, gat_Wlv + (size_t)i * W1c * W2c, t1, NN, W2c, W1c, stream);
    k_att<<<g1(NN * HEADS), 256, 0, stream>>>(t1, gat_aslv + i * W2c, a_s, NN, HEADS, H2);
    k_att<<<g1(NN * HEADS), 256, 0, stream>>>(t1, gat_adlv + i * W2c, a_d, NN, HEADS, H2);
    k_m_init<<<g1(NN * HEADS), 256, 0, stream>>>(a_s, a_d, mkey, NN * HEADS);
    k_m_edges<<<g1(NE * HEADS), 256, 0, stream>>>(a_s, a_d, src, dst, mkey, NE, HEADS);
    k_den_init<<<g1(NN * HEADS), 256, 0, stream>>>(a_s, a_d, mkey, den, NN * HEADS);
    k_den_edges<<<g1(NE * HEADS), 256, 0, stream>>>(a_s, a_d, src, dst, mkey, den, NE, HEADS);
    k_gat_agg_init<<<g1(NN * HEADS), 256, 0, stream>>>(t1, a_s, a_d, mkey, den, out2, NN, HEADS, H2);
    k_gat_agg_edges<<<g1(NE * HEADS), 256, 0, stream>>>(t1, a_s, a_d, mkey, den, src, dst, out2, NE, HEADS, H2);
    k_head_mean_bias<<<g1(NN * H2), 256, 0, stream>>>(out2, gat_blv + i * H2, lv, NN, HEADS, H2);

    k_combine<<<g1((long long)NN * H2), 256, 0, stream>>>(mu, lv, eps_gat + (size_t)i * NN * H2,
                                                          mu_acc, lv_acc, z_acc,
                                                          (long long)NN * H2, 0);
  }

  // ----------------- decoder: adj = sigmoid(z z^T) -----------------
  {
    int strips = (NN / 16) * (NN / 64);   // 65536 waves, 16x64 strip each
    int blocks = (strips + 7) / 8;
    k_decoder<<<dim3(blocks), dim3(256), 0, stream>>>(z_acc, out, NN, H2);
  }
  k_tail<<<g1(NN * H2), 256, 0, stream>>>(mu_acc, lv_acc, out, NN * H2);
}